// HyenaLayer_29661044146383
// MI455X (gfx1250) — compile-verified
//
#include <hip/hip_runtime.h>
#include <hip/hip_bf16.h>
#include <math.h>

// ---------------------------------------------------------------------------
// Hyena layer for MI455X (gfx1250), wave32.
//   B=2, L=8192, D=768, EMB=64, SHORT_K=3, FFT size 16384.
// GEMMs: v_wmma_f32_16x16x4_f32, 64x16 strip per wave (4x B-fragment reuse),
// software-pipelined fragment loads (next k-step loads before current WMMAs).
// Long conv: in-LDS radix-2 FFT (128KB LDS / workgroup; WGP has 320KB),
// rows staged into LDS via the Tensor Data Mover (tensor_load_to_lds).
// ---------------------------------------------------------------------------

#define D_MODEL 768
#define SEQ_LEN 8192
#define BATCH   2
#define EMB     64
#define NFFT    16384
#define LOG2N   14
#define M_ROWS  (BATCH * SEQ_LEN)     // 16384
#define N_QKV   (3 * D_MODEL)         // 2304
#define LDS_STAGE_BYTES 98304         // raw row staging: [98304, 131072)

typedef float v2f __attribute__((ext_vector_type(2)));
typedef float v8f __attribute__((ext_vector_type(8)));
typedef unsigned int u32x4 __attribute__((ext_vector_type(4)));
typedef int i32x4 __attribute__((ext_vector_type(4)));
typedef int i32x8 __attribute__((ext_vector_type(8)));

#if defined(__has_builtin)
# if __has_builtin(__builtin_amdgcn_tensor_load_to_lds) && \
     __has_builtin(__builtin_amdgcn_s_wait_tensorcnt)
#  define HAVE_TDM 1
# else
#  define HAVE_TDM 0
# endif
#else
# define HAVE_TDM 0
#endif

#define WMMA_F32(a, b, c) \
    __builtin_amdgcn_wmma_f32_16x16x4_f32(false, (a), false, (b), (short)0, (c), false, false)

// ---------------------------------------------------------------------------
// TDM: DMA one 8192-float row (32KB) from global into LDS at LDS_STAGE_BYTES.
// D# group0: count=1 | lds_addr | global_addr | type=2.
// D# group1: data_size=4B, tensor_dim0=tile_dim0=8192 (1-D tile), stride=8192.
// Issued by one wave; EXEC is ignored by tensor ops.
// ---------------------------------------------------------------------------
#if HAVE_TDM
__device__ __forceinline__ void tdm_load_row_8k(const float* gsrc) {
    const unsigned long long ga = (unsigned long long)(size_t)gsrc;
    u32x4 g0;
    g0[0] = 1u;                                              // count=1 (valid)
    g0[1] = (unsigned)LDS_STAGE_BYTES;                       // lds_addr
    g0[2] = (unsigned)(ga & 0xffffffffull);                  // global_addr lo
    g0[3] = (unsigned)((ga >> 32) & 0x01ffffffull)           // global_addr hi
          | (2u << 30);                                      // type=2 (image)
    i32x8 g1;
    g1[0] = (int)(2u << 16);          // workgroup_mask=0, data_size=2 (4B)
    g1[1] = (int)(8192u << 16);       // tensor_dim0[15:0] in [31:16]
    g1[2] = (int)(1u << 16);          // tensor_dim0 hi=0, tensor_dim1=1
    g1[3] = (int)(8192u << 16);       // tensor_dim1 hi=0, tile_dim0=8192
    g1[4] = 0;                        // tile_dim1=0, tile_dim2=0 (1-D tile)
    g1[5] = 8192;                     // tensor_dim0_stride lo
    g1[6] = 0;
    g1[7] = 0;
    const i32x4 gz = {0, 0, 0, 0};
# if __clang_major__ >= 23
    const i32x8 gz8 = {0, 0, 0, 0, 0, 0, 0, 0};
    __builtin_amdgcn_tensor_load_to_lds(g0, g1, gz, gz, gz8, 0);
# else
    __builtin_amdgcn_tensor_load_to_lds(g0, g1, gz, gz, 0);
# endif
}
#endif

// Stage one 8192-float row into smem[] as complex with zero imag + zero pad.
__device__ __forceinline__ void load_row_zero_pad(float2* smem, const float* gsrc,
                                                  int tid, int nthr) {
#if HAVE_TDM
    if (tid < 32) {                     // one wave issues the DMA
        tdm_load_row_8k(gsrc);
        __builtin_amdgcn_s_wait_tensorcnt(0);
    }
    __syncthreads();
    const float* raw = (const float*)((const char*)smem + LDS_STAGE_BYTES);
    for (int i = tid; i < SEQ_LEN; i += nthr)
        smem[i] = make_float2(raw[i], 0.0f);     // writes bytes < 65536: no clash
    __syncthreads();
#else
    for (int i = tid; i < SEQ_LEN; i += nthr)
        smem[i] = make_float2(gsrc[i], 0.0f);
#endif
    for (int i = SEQ_LEN + tid; i < NFFT; i += nthr)
        smem[i] = make_float2(0.0f, 0.0f);       // overwrites staging (dead now)
    __syncthreads();
}

// ---------------------------------------------------------------------------
// Software-pipelined 64x16-strip WMMA GEMM core. K = 768.
// a0..a3: A rows (4 M-tiles), br: B row (N-tile). Results in c0..c3.
// ---------------------------------------------------------------------------
__device__ __forceinline__ void gemm_strip_64x16(const float* a0, const float* a1,
                                                 const float* a2, const float* a3,
                                                 const float* br, int hi,
                                                 v8f& c0, v8f& c1, v8f& c2, v8f& c3) {
    // One shot of prefetches covering the 3KB rows (no branches in hot loop).
    __builtin_prefetch(a0 + 256, 0, 1);  __builtin_prefetch(a0 + 512, 0, 1);
    __builtin_prefetch(a1 + 256, 0, 1);  __builtin_prefetch(a1 + 512, 0, 1);
    __builtin_prefetch(a2 + 256, 0, 1);  __builtin_prefetch(a2 + 512, 0, 1);
    __builtin_prefetch(a3 + 256, 0, 1);  __builtin_prefetch(a3 + 512, 0, 1);
    __builtin_prefetch(br + 256, 0, 1);  __builtin_prefetch(br + 512, 0, 1);

    const int ko0 = 2 * hi;             // A/B 16x4: VGPR0=K+0, VGPR1=K+1
    v2f b_c  = *(const v2f*)(br + ko0);
    v2f av0c = *(const v2f*)(a0 + ko0);
    v2f av1c = *(const v2f*)(a1 + ko0);
    v2f av2c = *(const v2f*)(a2 + ko0);
    v2f av3c = *(const v2f*)(a3 + ko0);

    for (int kk = 0; kk < D_MODEL - 4; kk += 4) {
        const int kn = kk + 4 + 2 * hi;
        // Next k-step's fragments: issued before the current WMMAs so the
        // loads (one clause) overlap four matrix-op issues.
        v2f b_n  = *(const v2f*)(br + kn);
        v2f av0n = *(const v2f*)(a0 + kn);
        v2f av1n = *(const v2f*)(a1 + kn);
        v2f av2n = *(const v2f*)(a2 + kn);
        v2f av3n = *(const v2f*)(a3 + kn);
        c0 = WMMA_F32(av0c, b_c, c0);
        c1 = WMMA_F32(av1c, b_c, c1);
        c2 = WMMA_F32(av2c, b_c, c2);
        c3 = WMMA_F32(av3c, b_c, c3);
        b_c = b_n; av0c = av0n; av1c = av1n; av2c = av2n; av3c = av3n;
    }
    c0 = WMMA_F32(av0c, b_c, c0);       // drain
    c1 = WMMA_F32(av1c, b_c, c1);
    c2 = WMMA_F32(av2c, b_c, c2);
    c3 = WMMA_F32(av3c, b_c, c3);
}

// ---------------------------------------------------------------------------
// Kernel 1: qkv projection GEMM.  C[M,N] = X[M,K] * W[N,K]^T  (K = 768)
// Epilogue routes N-columns into Q [B,L,D] and transposed Kt/Vt [B,D,L].
// ---------------------------------------------------------------------------
__global__ void proj_gemm(const float* __restrict__ X, const float* __restrict__ W,
                          float* __restrict__ Q, float* __restrict__ Kt,
                          float* __restrict__ Vt) {
    const int lane = threadIdx.x & 31;
    const int wave = threadIdx.x >> 5;
    const int lo = lane & 15;           // M index (A) / N index (B,C)
    const int hi = lane >> 4;           // lane-half selector
    const int m0 = blockIdx.x * 64;
    const int n0 = (blockIdx.y * 8 + wave) * 16;

    const float* a0 = X + (size_t)(m0 + 0  + lo) * D_MODEL;
    const float* a1 = X + (size_t)(m0 + 16 + lo) * D_MODEL;
    const float* a2 = X + (size_t)(m0 + 32 + lo) * D_MODEL;
    const float* a3 = X + (size_t)(m0 + 48 + lo) * D_MODEL;
    const float* br = W + (size_t)(n0 + lo) * D_MODEL;

    v8f c0 = {}, c1 = {}, c2 = {}, c3 = {};
    gemm_strip_64x16(a0, a1, a2, a3, br, hi, c0, c1, c2, c3);

    const int n = n0 + lo;
    #pragma unroll
    for (int t = 0; t < 4; ++t) {
        const v8f c = (t == 0) ? c0 : (t == 1) ? c1 : (t == 2) ? c2 : c3;
        #pragma unroll
        for (int v = 0; v < 8; ++v) {
            const int m  = m0 + 16 * t + v + 8 * hi;   // C: VGPR v -> M=v / v+8
            const int b_ = m >> 13;                    // / SEQ_LEN
            const int l  = m & (SEQ_LEN - 1);
            const float val = c[v];
            if (n < D_MODEL) {
                Q[(size_t)m * D_MODEL + n] = val;
            } else if (n < 2 * D_MODEL) {
                Kt[((size_t)b_ * D_MODEL + (n - D_MODEL)) * SEQ_LEN + l] = val;
            } else {
                Vt[((size_t)b_ * D_MODEL + (n - 2 * D_MODEL)) * SEQ_LEN + l] = val;
            }
        }
    }
}

// ---------------------------------------------------------------------------
// Kernel 2: causal depthwise conv (k=3) on k and v, then elementwise product.
// y[t] = w0*u[t-2] + w1*u[t-1] + w2*u[t] + b
// ---------------------------------------------------------------------------
__global__ void dwconv_mul(const float* __restrict__ Kt, const float* __restrict__ Vt,
                           const float* __restrict__ kw, const float* __restrict__ kb,
                           const float* __restrict__ vw, const float* __restrict__ vb,
                           float* __restrict__ KV) {
    const size_t idx = (size_t)blockIdx.x * blockDim.x + threadIdx.x;
    const size_t total = (size_t)BATCH * D_MODEL * SEQ_LEN;
    if (idx >= total) return;
    const int l = (int)(idx & (SEQ_LEN - 1));
    const int d = (int)((idx >> 13) % D_MODEL);

    const float k2 = Kt[idx];
    const float k1 = (l >= 1) ? Kt[idx - 1] : 0.0f;
    const float k0 = (l >= 2) ? Kt[idx - 2] : 0.0f;
    const float kc = kw[d * 3 + 0] * k0 + kw[d * 3 + 1] * k1 + kw[d * 3 + 2] * k2 + kb[d];

    const float v2 = Vt[idx];
    const float v1 = (l >= 1) ? Vt[idx - 1] : 0.0f;
    const float v0 = (l >= 2) ? Vt[idx - 2] : 0.0f;
    const float vc = vw[d * 3 + 0] * v0 + vw[d * 3 + 1] * v1 + vw[d * 3 + 2] * v2 + vb[d];

    KV[idx] = kc * vc;
}

// ---------------------------------------------------------------------------
// Kernel 3: hyena filter MLP + decay -> time-domain filter Ht [D, L]
// (FFT zero-padding synthesized later in LDS). One 64-thread block per l.
// ---------------------------------------------------------------------------
__global__ void filter_mlp(const float* __restrict__ W1, const float* __restrict__ b1,
                           const float* __restrict__ W2, const float* __restrict__ b2,
                           const float* __restrict__ W3, const float* __restrict__ b3,
                           const float* __restrict__ logdec, float* __restrict__ Ht) {
    __shared__ float pe[EMB], h1[EMB], h2[EMB];
    const int l = blockIdx.x;           // 0..L-1
    const int j = threadIdx.x;          // 0..63

    const float t = (float)l / (float)(SEQ_LEN - 1);      // linspace(0,1,L)
    {
        const float f = 6.28318530717958647f * (float)((j & 31) + 1);
        pe[j] = (j < 32) ? __sinf(t * f) : __cosf(t * f);
    }
    __syncthreads();

    float acc = b1[j];
    #pragma unroll 8
    for (int i = 0; i < EMB; ++i) acc += pe[i] * W1[j * EMB + i];
    h1[j] = acc / (1.0f + __expf(-acc));                  // silu
    __syncthreads();

    acc = b2[j];
    #pragma unroll 8
    for (int i = 0; i < EMB; ++i) acc += h1[i] * W2[j * EMB + i];
    h2[j] = acc / (1.0f + __expf(-acc));
    __syncthreads();

    for (int d = j; d < D_MODEL; d += EMB) {
        float a = b3[d];
        #pragma unroll 8
        for (int i = 0; i < EMB; ++i) a += h2[i] * W3[d * EMB + i];
        const float dec = __expf(-fabsf(logdec[d]) * (float)l);
        Ht[(size_t)d * SEQ_LEN + l] = a * dec;
    }
}

// ---------------------------------------------------------------------------
// In-LDS iterative radix-2 FFT, size 16384 (128 KB of LDS as float2).
// sign = -1: forward DFT; sign = +1: inverse kernel (caller scales by 1/N).
// ---------------------------------------------------------------------------
__device__ __forceinline__ void fft_lds(float2* s, int tid, int nthr, float sign) {
    // bit-reversal permutation
    for (int i = tid; i < NFFT; i += nthr) {
        const int j = (int)(__brev((unsigned)i) >> (32 - LOG2N));
        if (j > i) { float2 tmp = s[i]; s[i] = s[j]; s[j] = tmp; }
    }
    __syncthreads();
    for (int st = 0; st < LOG2N; ++st) {
        const int   half = 1 << st;
        const float ang  = sign * 6.28318530717958647f / (float)(half << 1);
        for (int bf = tid; bf < (NFFT >> 1); bf += nthr) {
            const int pos = bf & (half - 1);
            const int i   = ((bf >> st) << (st + 1)) + pos;
            const int k   = i + half;
            float sa, ca;
            __sincosf(ang * (float)pos, &sa, &ca);
            const float2 u = s[i];
            const float2 v = s[k];
            const float vr = v.x * ca - v.y * sa;
            const float vi = v.x * sa + v.y * ca;
            s[i] = make_float2(u.x + vr, u.y + vi);
            s[k] = make_float2(u.x - vr, u.y - vi);
        }
        __syncthreads();
    }
}

// Kernel 4: forward FFT of each filter row -> Hf [D, NFFT] complex
__global__ void fft_filter(const float* __restrict__ Ht, float2* __restrict__ Hf) {
    extern __shared__ float2 smem[];
    const int d = blockIdx.x, tid = threadIdx.x, nthr = blockDim.x;
    load_row_zero_pad(smem, Ht + (size_t)d * SEQ_LEN, tid, nthr);
    fft_lds(smem, tid, nthr, -1.0f);
    float2* dst = Hf + (size_t)d * NFFT;
    for (int i = tid; i < NFFT; i += nthr) dst[i] = smem[i];
}

// Kernel 5: per-(b,d) row: FFT(kv), * Hf[d], IFFT, take real[:L], gate with
// silu(q), write gated Y in [B,L,D] layout for the output GEMM.
__global__ void fft_conv_gate(const float* __restrict__ KV, const float2* __restrict__ Hf,
                              const float* __restrict__ Q, float* __restrict__ Yg) {
    extern __shared__ float2 smem[];
    const int row = blockIdx.x;                 // 0 .. B*D-1
    const int b = row / D_MODEL, d = row % D_MODEL;
    const int tid = threadIdx.x, nthr = blockDim.x;

    load_row_zero_pad(smem, KV + (size_t)row * SEQ_LEN, tid, nthr);

    fft_lds(smem, tid, nthr, -1.0f);            // forward

    const float2* hf = Hf + (size_t)d * NFFT;
    for (int i = tid; i < NFFT; i += nthr) {    // pointwise complex multiply
        const float2 a = smem[i], h = hf[i];
        smem[i] = make_float2(a.x * h.x - a.y * h.y, a.x * h.y + a.y * h.x);
    }
    __syncthreads();

    fft_lds(smem, tid, nthr, +1.0f);            // inverse (scale below)

    const float inv = 1.0f / (float)NFFT;
    for (int l = tid; l < SEQ_LEN; l += nthr) {
        const float y = smem[l].x * inv;
        const float q = Q[((size_t)b * SEQ_LEN + l) * D_MODEL + d];
        const float g = q / (1.0f + __expf(-q));          // silu gate
        Yg[((size_t)b * SEQ_LEN + l) * D_MODEL + d] = g * y;
    }
}

// ---------------------------------------------------------------------------
// Kernel 6: output projection GEMM.  out[M,768] = Yg[M,768] * Wout[768,768]^T
// ---------------------------------------------------------------------------
__global__ void out_gemm(const float* __restrict__ A, const float* __restrict__ W,
                         float* __restrict__ out) {
    const int lane = threadIdx.x & 31;
    const int wave = threadIdx.x >> 5;
    const int lo = lane & 15;
    const int hi = lane >> 4;
    const int m0 = blockIdx.x * 64;
    const int n0 = (blockIdx.y * 8 + wave) * 16;

    const float* a0 = A + (size_t)(m0 + 0  + lo) * D_MODEL;
    const float* a1 = A + (size_t)(m0 + 16 + lo) * D_MODEL;
    const float* a2 = A + (size_t)(m0 + 32 + lo) * D_MODEL;
    const float* a3 = A + (size_t)(m0 + 48 + lo) * D_MODEL;
    const float* br = W + (size_t)(n0 + lo) * D_MODEL;

    v8f c0 = {}, c1 = {}, c2 = {}, c3 = {};
    gemm_strip_64x16(a0, a1, a2, a3, br, hi, c0, c1, c2, c3);

    const int n = n0 + lo;
    #pragma unroll
    for (int t = 0; t < 4; ++t) {
        const v8f c = (t == 0) ? c0 : (t == 1) ? c1 : (t == 2) ? c2 : c3;
        #pragma unroll
        for (int v = 0; v < 8; ++v) {
            const int m = m0 + 16 * t + v + 8 * hi;
            out[(size_t)m * D_MODEL + n] = c[v];
        }
    }
}

// ---------------------------------------------------------------------------
// Host launcher
// ---------------------------------------------------------------------------
extern "C" void kernel_launch(void* const* d_in, const int* in_sizes, int n_in,
                              void* d_out, int out_size, void* d_ws, size_t ws_size,
                              hipStream_t stream) {
    const float* x        = (const float*)d_in[0];
    const float* in_w     = (const float*)d_in[1];
    const float* sck_w    = (const float*)d_in[2];
    const float* sck_b    = (const float*)d_in[3];
    const float* scv_w    = (const float*)d_in[4];
    const float* scv_b    = (const float*)d_in[5];
    const float* mlp_w1   = (const float*)d_in[6];
    const float* mlp_b1   = (const float*)d_in[7];
    const float* mlp_w2   = (const float*)d_in[8];
    const float* mlp_b2   = (const float*)d_in[9];
    const float* mlp_w3   = (const float*)d_in[10];
    const float* mlp_b3   = (const float*)d_in[11];
    const float* log_dec  = (const float*)d_in[12];
    const float* out_w    = (const float*)d_in[13];
    float* out = (float*)d_out;

    // Workspace layout (bytes), with lifetime-based aliasing:
    //   [0,           50331648)  Q   [B,L,D]            (live to kernel 5)
    //   [50331648,   150994944)  Kt|Vt (k1..k2) then Hf (k4..k5)
    //   [150994944,  201326592)  KV  [B,D,L]            (k2..k5)
    //   [201326592,  251658240)  Ht [D,L] (k3..k4) then Yg (k5..k6)
    char* ws = (char*)d_ws;
    float*  Q    = (float*) (ws + 0);
    float*  Kt   = (float*) (ws + 50331648);
    float*  Vt   = (float*) (ws + 100663296);
    float2* Hf   = (float2*)(ws + 50331648);     // aliases Kt|Vt (dead after k2)
    float*  KV   = (float*) (ws + 150994944);
    float*  Ht   = (float*) (ws + 201326592);
    float*  Yg   = (float*) (ws + 201326592);    // aliases Ht (dead after k4)
    (void)ws_size; (void)in_sizes; (void)n_in; (void)out_size;

    // 1) qkv projection (WMMA): M=16384, N=2304, K=768; 64x16 strip per wave
    proj_gemm<<<dim3(M_ROWS / 64, N_QKV / (16 * 8)), 256, 0, stream>>>(
        x, in_w, Q, Kt, Vt);

    // 2) depthwise causal conv + product
    {
        const size_t total = (size_t)BATCH * D_MODEL * SEQ_LEN;
        dwconv_mul<<<(unsigned)((total + 255) / 256), 256, 0, stream>>>(
            Kt, Vt, sck_w, sck_b, scv_w, scv_b, KV);
    }

    // 3) filter MLP + decay -> Ht [D, L]
    filter_mlp<<<SEQ_LEN, EMB, 0, stream>>>(mlp_w1, mlp_b1, mlp_w2, mlp_b2,
                                            mlp_w3, mlp_b3, log_dec, Ht);

    // 4) FFT of filter rows -> Hf (TDM row staging; 128 KB LDS each)
    fft_filter<<<D_MODEL, 256, NFFT * sizeof(float2), stream>>>(Ht, Hf);

    // 5) FFT conv + silu(q) gating -> Yg [B,L,D]
    fft_conv_gate<<<BATCH * D_MODEL, 256, NFFT * sizeof(float2), stream>>>(
        KV, Hf, Q, Yg);

    // 6) output projection (WMMA): M=16384, N=768, K=768
    out_gemm<<<dim3(M_ROWS / 64, D_MODEL / (16 * 8)), 256, 0, stream>>>(
        Yg, out_w, out);
}